// HO_27333171871824
// MI455X (gfx1250) — compile-verified
//
#include <hip/hip_runtime.h>
#include <math.h>

#define NTOT 8192
#define NT1  4278
#define NT2  2500
#define NT3  1414
#define DH   64
#define DRAW 256

typedef __attribute__((ext_vector_type(2))) float v2f;
typedef __attribute__((ext_vector_type(8))) float v8f;

__device__ __forceinline__ float elu_f(float x) {
  return x > 0.0f ? x : (expf(x) - 1.0f);
}

// ---------------------------------------------------------------------------
// h[row, col] = f_t[row_local, :] @ w_t[:, col] + b_t[col]   (per node type)
// ---------------------------------------------------------------------------
__global__ __launch_bounds__(256) void hgcn_proj_kernel(
    const float* __restrict__ f0, const float* __restrict__ f1,
    const float* __restrict__ f2, const float* __restrict__ w0,
    const float* __restrict__ b0, const float* __restrict__ w1,
    const float* __restrict__ b1, const float* __restrict__ w2,
    const float* __restrict__ b2, float* __restrict__ h) {
  int idx = blockIdx.x * 256 + threadIdx.x;   // 0 .. N*DH-1
  int row = idx >> 6;
  int col = idx & 63;
  const float* f; const float* w; const float* b; int r;
  if (row < NT1)              { f = f0; w = w0; b = b0; r = row; }
  else if (row < NT1 + NT2)   { f = f1; w = w1; b = b1; r = row - NT1; }
  else                        { f = f2; w = w2; b = b2; r = row - NT1 - NT2; }
  const float* fr = f + (size_t)r * DRAW;
  float acc = b[col];
  #pragma unroll 8
  for (int k = 0; k < DRAW; ++k) acc += fr[k] * w[k * DH + col];
  h[idx] = acc;
}

// ---------------------------------------------------------------------------
// g[row, :] = beta0*mate0 + beta1*mate1 + labeladj (raw, un-normalized)
// rinv[row] = 1 / max(||g[row,:]||_2, 1e-12)
// ---------------------------------------------------------------------------
__global__ __launch_bounds__(256) void hgcn_build_g_kernel(
    const float* __restrict__ labeladj, const float* __restrict__ mate0,
    const float* __restrict__ mate1, const float* __restrict__ beta,
    float* __restrict__ g, float* __restrict__ rinv) {
  const int row = blockIdx.x;
  const float be0 = beta[0], be1 = beta[1];
  const size_t off = (size_t)row * NTOT;
  float ss = 0.0f;
  for (int j = threadIdx.x; j < NTOT; j += 256) {
    float v = be0 * mate0[off + j] + be1 * mate1[off + j] + labeladj[off + j];
    g[off + j] = v;
    ss += v * v;
  }
  __shared__ float red[256];
  red[threadIdx.x] = ss;
  __syncthreads();
  for (int s = 128; s > 0; s >>= 1) {
    if (threadIdx.x < s) red[threadIdx.x] += red[threadIdx.x + s];
    __syncthreads();
  }
  if (threadIdx.x == 0) {
    float nrm = sqrtf(red[0]);
    rinv[row] = 1.0f / fmaxf(nrm, 1e-12f);
  }
}

// ---------------------------------------------------------------------------
// support[row, col] = h[row, :] @ w[:, col]     (8192x64 @ 64x64, tiny)
// ---------------------------------------------------------------------------
__global__ __launch_bounds__(256) void hgcn_xform_kernel(
    const float* __restrict__ h, const float* __restrict__ w,
    float* __restrict__ s) {
  int idx = blockIdx.x * 256 + threadIdx.x;
  int row = idx >> 6;
  int col = idx & 63;
  const float* hr = h + (size_t)row * DH;
  float acc = 0.0f;
  #pragma unroll 8
  for (int k = 0; k < DH; ++k) acc += hr[k] * w[k * DH + col];
  s[idx] = acc;
}

// ---------------------------------------------------------------------------
// out = elu(rinv[row] * (g_raw @ support) + bias)  via V_WMMA_F32_16X16X4_F32
// Wave -> 16 rows x 32 cols (two 16x16 accumulators).
// Block = 4 waves = 32 rows x 64 cols; g streams from HBM exactly once.
// ---------------------------------------------------------------------------
__global__ __launch_bounds__(128) void hgcn_agg_kernel(
    const float* __restrict__ g, const float* __restrict__ rinv,
    const float* __restrict__ s, const float* __restrict__ bias,
    float* __restrict__ out, int row_limit) {
  const int lane = threadIdx.x & 31;
  const int wave = threadIdx.x >> 5;
  const int rbase = blockIdx.x * 32 + (wave >> 1) * 16;  // 16-row tile
  const int nbase = (wave & 1) * 32;                     // 32-col half
  const int m  = lane & 15;            // row (A) / col (B) within tile
  const int kh = (lane >> 4) << 1;     // k sub-offset: 0 or 2

  const float* grow = g + (size_t)(rbase + m) * NTOT + kh;  // A fragment base
  const float* sp   = s + nbase + m;                        // B fragment base

  v8f acc0 = {};
  v8f acc1 = {};

  for (int k = 0; k < NTOT; k += 8) {
    #pragma unroll
    for (int u = 0; u < 2; ++u) {
      const int kk = k + u * 4;
      v2f a;
      a.x = grow[kk];          // A[m][kk+kh]
      a.y = grow[kk + 1];      // A[m][kk+kh+1]  (single b64 load)
      const int r0 = (kk + kh) * DH;
      v2f b0, b1;
      b0.x = sp[r0];           // B[kh  ][n],   cols nbase..nbase+15
      b0.y = sp[r0 + DH];      // B[kh+1][n]
      b1.x = sp[r0 + 16];      // cols nbase+16..nbase+31
      b1.y = sp[r0 + DH + 16];
      acc0 = __builtin_amdgcn_wmma_f32_16x16x4_f32(
          false, a, false, b0, (short)0, acc0, false, false);
      acc1 = __builtin_amdgcn_wmma_f32_16x16x4_f32(
          false, a, false, b1, (short)0, acc1, false, false);
    }
  }

  const float bias0 = bias[nbase + m];
  const float bias1 = bias[nbase + 16 + m];
  #pragma unroll
  for (int v = 0; v < 8; ++v) {
    int row = rbase + (lane >> 4) * 8 + v;
    if (row < row_limit) {
      float rv = rinv[row];
      float x0 = elu_f(rv * acc0[v] + bias0);
      float x1 = elu_f(rv * acc1[v] + bias1);
      out[(size_t)row * DH + nbase + m]      = x0;
      out[(size_t)row * DH + nbase + 16 + m] = x1;
    }
  }
}

// ---------------------------------------------------------------------------
extern "C" void kernel_launch(void* const* d_in, const int* in_sizes, int n_in,
                              void* d_out, int out_size, void* d_ws,
                              size_t ws_size, hipStream_t stream) {
  const float* labeladj = (const float*)d_in[0];
  const float* mate0    = (const float*)d_in[1];
  const float* mate1    = (const float*)d_in[2];
  const float* f0       = (const float*)d_in[3];
  const float* f1       = (const float*)d_in[4];
  const float* f2       = (const float*)d_in[5];
  const float* beta     = (const float*)d_in[6];
  const float* w0       = (const float*)d_in[7];
  const float* b0       = (const float*)d_in[8];
  const float* w1       = (const float*)d_in[9];
  const float* b1       = (const float*)d_in[10];
  const float* w2       = (const float*)d_in[11];
  const float* b2       = (const float*)d_in[12];
  const float* gw0      = (const float*)d_in[13];
  const float* gb0      = (const float*)d_in[14];
  const float* gw1      = (const float*)d_in[15];
  const float* gb1      = (const float*)d_in[16];

  float* ws   = (float*)d_ws;
  float* g    = ws;                               // N*N
  float* rinv = g + (size_t)NTOT * NTOT;          // N
  float* h    = rinv + NTOT;                      // N*DH
  float* sup  = h + (size_t)NTOT * DH;            // N*DH
  float* h2   = sup + (size_t)NTOT * DH;          // N*DH

  const int elemBlocks = (NTOT * DH) / 256;       // 2048
  const int aggBlocks  = NTOT / 32;               // 256

  hgcn_proj_kernel<<<elemBlocks, 256, 0, stream>>>(f0, f1, f2, w0, b0, w1, b1,
                                                   w2, b2, h);
  hgcn_build_g_kernel<<<NTOT, 256, 0, stream>>>(labeladj, mate0, mate1, beta,
                                                g, rinv);
  // layer 1: gw0/gb0
  hgcn_xform_kernel<<<elemBlocks, 256, 0, stream>>>(h, gw0, sup);
  hgcn_agg_kernel<<<aggBlocks, 128, 0, stream>>>(g, rinv, sup, gb0, h2, NTOT);
  // layer 2: gw1/gb1
  hgcn_xform_kernel<<<elemBlocks, 256, 0, stream>>>(h2, gw1, sup);
  hgcn_agg_kernel<<<aggBlocks, 128, 0, stream>>>(g, rinv, sup, gb1, h, NTOT);
  // layer 3: gw1/gb1 again, write rows < 4278 straight to d_out
  hgcn_xform_kernel<<<elemBlocks, 256, 0, stream>>>(h, gw1, sup);
  hgcn_agg_kernel<<<aggBlocks, 128, 0, stream>>>(g, rinv, sup, gb1,
                                                 (float*)d_out, NT1);
}